// PDistLayer_91001767067820
// MI455X (gfx1250) — compile-verified
//
#include <hip/hip_runtime.h>

// PDist: out[u*(u-1)/2 - 1 + l] = ||x_l||^2 + ||x_u||^2 - 2<x_l,x_u>, u>=2, l<u
// N=8192 rows, D=128. fp32 throughout; Gram tiles via V_WMMA_F32_16X16X4_F32.

#define NROWS 8192
#define DIM   128

typedef __attribute__((ext_vector_type(2))) float v2f;
typedef __attribute__((ext_vector_type(8))) float v8f;

// ---------------- Kernel 1: row squared norms + trailing zero ----------------
__global__ void pdist_sqnorm_kernel(const float* __restrict__ x,
                                    float* __restrict__ sq,
                                    float* __restrict__ out,
                                    long long total) {
    int i = blockIdx.x * blockDim.x + threadIdx.x;
    if (i < NROWS) {
        const float4* row = (const float4*)(x + (size_t)i * DIM);
        float acc = 0.0f;
#pragma unroll
        for (int j = 0; j < DIM / 4; ++j) {
            float4 v = row[j];
            acc = fmaf(v.x, v.x, acc);
            acc = fmaf(v.y, v.y, acc);
            acc = fmaf(v.z, v.z, acc);
            acc = fmaf(v.w, v.w, acc);
        }
        sq[i] = acc;
    }
    if (i == 0) {
        // pair (0,1) is skipped by the reference loop; last stacked slot is 0
        out[total - 1] = 0.0f;
    }
}

// ---------------- Kernel 2: one wave32 per 16x16 Gram tile ----------------
// A-matrix 16x4 f32 layout (ISA 7.12.2): lanes 0-15 hold M=lane, K={0,1} in
// VGPR0/1; lanes 16-31 hold M=lane-16, K={2,3}. B=X^T for the Gram product,
// so B's per-lane fetch pattern is identical with the column-tile row base.
// C/D layout: lane L, elem j -> M = j + 8*(L>>4), N = L&15.
__global__ void __launch_bounds__(32)
pdist_tile_kernel(const float* __restrict__ x,
                  const float* __restrict__ sq,
                  float* __restrict__ out) {
    const int tu = blockIdx.x;   // column tile (u)
    const int tl = blockIdx.y;   // row tile (l)
    if (tu < tl) return;         // uniform exit: EXEC all-1s for surviving waves

    const int lane = threadIdx.x;        // 0..31
    const int lm   = lane & 15;
    const int kg   = (lane >> 4) << 1;   // 0 or 2: K sub-offset for this lane group

    const float* __restrict__ arow = x + (size_t)(tl * 16 + lm) * DIM + kg;
    const float* __restrict__ brow = x + (size_t)(tu * 16 + lm) * DIM + kg;

    v8f c = {};
#pragma unroll
    for (int s = 0; s < DIM / 4; ++s) {
        v2f a = *(const v2f*)(arow + 4 * s);   // global_load_b64, L2-resident
        v2f b = *(const v2f*)(brow + 4 * s);
        // D = A x B + C ; 8 args: (neg_a, A, neg_b, B, c_mod, C, reuse_a, reuse_b)
        c = __builtin_amdgcn_wmma_f32_16x16x4_f32(
                false, a, false, b, (short)0, c, false, false);
    }

    // Epilogue: dist = sq[l] + sq[u] - 2*gram. Each lane owns column u and 8
    // consecutive rows l -> 8 consecutive output slots (idx = u(u-1)/2-1+l).
    const int u     = tu * 16 + lm;
    const int mbase = tl * 16 + ((lane >> 4) << 3);
    const float squ = sq[u];
    const long long ubase = (long long)u * (u - 1) / 2 - 1;

    if (tu > tl) {
        // Off-diagonal tile: every element satisfies l < u (and u >= 16 >= 2).
        float* __restrict__ o = out + (ubase + mbase);
#pragma unroll
        for (int j = 0; j < 8; ++j) {
            o[j] = sq[mbase + j] + squ - 2.0f * c[j];
        }
    } else {
        // Diagonal tile: guard l < u (u >= 2 holds whenever l < u except (0,1)).
#pragma unroll
        for (int j = 0; j < 8; ++j) {
            const int l = mbase + j;
            if (l < u && u >= 2) {
                out[ubase + l] = sq[l] + squ - 2.0f * c[j];
            }
        }
    }
}

// ---------------- Launch ----------------
extern "C" void kernel_launch(void* const* d_in, const int* in_sizes, int n_in,
                              void* d_out, int out_size, void* d_ws, size_t ws_size,
                              hipStream_t stream) {
    (void)in_sizes; (void)n_in; (void)out_size; (void)ws_size;

    const float* x = (const float*)d_in[0];
    float* out = (float*)d_out;
    float* sq  = (float*)d_ws;   // NROWS floats of scratch

    const long long total = (long long)NROWS * (NROWS - 1) / 2;

    pdist_sqnorm_kernel<<<(NROWS + 127) / 128, 128, 0, stream>>>(x, sq, out, total);

    dim3 grid(NROWS / 16, NROWS / 16);   // 512 x 512 tiles, upper triangle does work
    pdist_tile_kernel<<<grid, 32, 0, stream>>>(x, sq, out);
}